// SynthesisBlock_24773371363996
// MI455X (gfx1250) — compile-verified
//
#include <hip/hip_runtime.h>
#include <hip/hip_bf16.h>
#include <math.h>
#include <stdint.h>

typedef __attribute__((ext_vector_type(2))) float    v2f;
typedef __attribute__((ext_vector_type(8))) float    v8f;
typedef __attribute__((ext_vector_type(4))) uint32_t u32x4;
typedef __attribute__((ext_vector_type(8))) uint32_t u32x8;

#define INV_SQRT512 0.044194173824159216f
#define LRELU_GAIN  1.41421356237309515f

// Tensor Data Mover staging of the weight slab (fallback: per-thread loads)
#define USE_TDM 1

// ---------------------------------------------------------------------------
// s = (ws[:,j] @ aw.T) / sqrt(512) + ab, then * post   -> (8,512)
// ---------------------------------------------------------------------------
__global__ void affine_kernel(const float* __restrict__ stylev, int j,
                              const float* __restrict__ aw,
                              const float* __restrict__ ab,
                              float post, float* __restrict__ sOut)
{
    int idx = blockIdx.x * 256 + threadIdx.x;   // 0..4095
    int b = idx >> 9, o = idx & 511;
    const float* wrow = stylev + (size_t)(b * 3 + j) * 512;
    const float* arow = aw + (size_t)o * 512;
    float acc = 0.f;
    for (int k = 0; k < 512; ++k) acc += wrow[k] * arow[k];
    sOut[idx] = (acc * INV_SQRT512 + ab[o]) * post;
}

// wsq[o,i] = sum over 9 taps of w[o,i,:,:]^2
__global__ void wsq_kernel(const float* __restrict__ w, float* __restrict__ wsq)
{
    int idx = blockIdx.x * 256 + threadIdx.x;   // 0..262143
    const float* p = w + (size_t)idx * 9;
    float a = 0.f;
    #pragma unroll
    for (int t = 0; t < 9; ++t) a += p[t] * p[t];
    wsq[idx] = a;
}

// d[b,o] = rsqrt(sum_i s[b,i]^2 * wsq[o,i] + 1e-8)
__global__ void demod_kernel(const float* __restrict__ s,
                             const float* __restrict__ wsq,
                             float* __restrict__ d)
{
    int idx = blockIdx.x * 256 + threadIdx.x;   // 0..4095
    int b = idx >> 9, o = idx & 511;
    const float* sp = s + (size_t)b * 512;
    const float* wp = wsq + (size_t)o * 512;
    float acc = 1e-8f;
    for (int i = 0; i < 512; ++i) { float sv = sp[i]; acc += sv * sv * wp[i]; }
    d[idx] = 1.0f / sqrtf(acc);
}

// repack (512,512,3,3) -> (9,512,512) : wpk[tap][o][i] = w[o][i][tap]
// makes the per-(tap,ichunk) WMMA A-slab a contiguous-row 2D tile for the TDM
__global__ void pack_w_kernel(const float* __restrict__ w, float* __restrict__ wpk)
{
    int idx = blockIdx.x * 256 + threadIdx.x;   // 0..2359295
    int t = idx / 262144;
    int r = idx - t * 262144;                   // o*512+i
    wpk[(size_t)t * 262144 + r] = w[(size_t)r * 9 + t];
}

// ---------------------------------------------------------------------------
// Implicit-GEMM modulated conv with fp32 WMMA.
//   UP=true : conv0 — input x (8,512,32,32); the up=2 / pad(3,2) / gain 4
//             4-tap upfirdn is fused into the halo stage (66x66 virtual
//             input), and the weight spatial flip folds into tap -> 8-tap.
//   UP=false: conv1 — input h0 (8,512,64,64), SAME pad 1, no flip.
// Block: 256 thr = 8 waves, tile M=128 ochans x N=64 pixels (8x8 spatial).
// Wave (wm,wn) in 4x2 grid owns 32x32 = 2x2 of 16x16 WMMA accumulators.
// K loop: 16-input-channel chunks x 9 taps x 4 wmma-K-steps (K=4 each).
// Weight slab staged LDS-side by the Tensor Data Mover (TENSORcnt).
// ---------------------------------------------------------------------------
template<bool UP>
__global__ __launch_bounds__(256)
void conv_wmma_kernel(const float* __restrict__ in,
                      const float* __restrict__ wpk,    // (9,512,512) packed
                      const float* __restrict__ sIn,    // (8,512)
                      const float* __restrict__ dmod,   // (8,512)
                      const float* __restrict__ noise,  // (64,64)
                      const float* __restrict__ nsPtr,  // scalar
                      const float* __restrict__ bias,   // (512)
                      float* __restrict__ out)          // (8,512,64,64)
{
    __shared__ float Ah[128 * 17];        // weights: 128 ochans x 16 ich (pad 17)
    __shared__ float Bh[16 * 10 * 11];    // halo: 16 ichans x 10 x 10 (pad 11)

    const int tid   = threadIdx.x;
    const int lane  = tid & 31;
    const int wave  = tid >> 5;
    const int wm    = wave >> 1;          // 0..3  (M)
    const int wn    = wave & 1;           // 0..1  (N)
    const int tileY = (int)blockIdx.x >> 3;
    const int tileX = (int)blockIdx.x & 7;
    const int oBase = (int)blockIdx.y * 128;
    const int b     = (int)blockIdx.z;

    // fp32 WMMA fragment geometry (ISA 7.12.2):
    //   A(m,k): lane = (k>=2)*16 + m, reg = k&1 ; B(k,n): lane = (k>=2)*16 + n
    const int mrow  = lane & 15;
    const int khalf = (lane >> 4) * 2;
    const int n0    = wn * 32 + (lane & 15);
    const int n1    = n0 + 16;
    const int py0 = n0 >> 3, px0 = n0 & 7;
    const int py1 = n1 >> 3, px1 = n1 & 7;

    v8f c00 = {}, c01 = {}, c10 = {}, c11 = {};

    for (int icB = 0; icB < 512; icB += 16) {
        __syncthreads();
        // stage style-scaled input halo: 16 ch x 10 x 10
        for (int idx = tid; idx < 1600; idx += 256) {
            int cch = idx / 100;
            int r   = idx - cch * 100;
            int hy  = r / 10, hx = r - hy * 10;
            float v;
            if (UP) {
                // fused upfirdn2d(up=2, pad0=3, pad1=2, gain=4) at (gy,gx) in 66x66
                const float F4[4] = {0.125f, 0.375f, 0.375f, 0.125f};
                const float* plane = in + ((size_t)b * 512 + icB + cch) * 1024;
                int gy = tileY * 8 + hy;
                int gx = tileX * 8 + hx;
                float acc = 0.f;
                #pragma unroll
                for (int ty = 0; ty < 4; ++ty) {
                    int zr = gy + ty - 3;
                    if (zr < 0 || zr >= 64 || (zr & 1)) continue;
                    int ir = zr >> 1;
                    #pragma unroll
                    for (int tx = 0; tx < 4; ++tx) {
                        int zc = gx + tx - 3;
                        if (zc < 0 || zc >= 64 || (zc & 1)) continue;
                        acc += F4[ty] * F4[tx] * 4.0f * plane[ir * 32 + (zc >> 1)];
                    }
                }
                v = acc;
            } else {
                int gy = tileY * 8 + hy - 1;
                int gx = tileX * 8 + hx - 1;
                v = 0.f;
                if (gy >= 0 && gy < 64 && gx >= 0 && gx < 64)
                    v = in[(((size_t)b * 512 + icB + cch) * 64 + gy) * 64 + gx];
            }
            v *= sIn[b * 512 + icB + cch];
            Bh[(cch * 10 + hy) * 11 + hx] = v;
        }
        for (int tap = 0; tap < 9; ++tap) {
            int tapf = UP ? 8 - tap : tap;          // spatial flip for up-conv
            int ky   = tap / 3, kx = tap - ky * 3;
            __syncthreads();                        // prior Ah readers done
#if USE_TDM
            if (wave == 0) {
                // Tensor DMA: 2D tile, 128 rows x 16 f32, row stride 512 f32,
                // LDS pad 1 DWORD per 16 DWORDs -> 17-float LDS row stride.
                const float* gsrc = wpk + ((size_t)tapf * 512 + oBase) * 512 + icB;
                uint64_t ga   = (uint64_t)(uintptr_t)gsrc;
                uint32_t ldsa = (uint32_t)(uintptr_t)(&Ah[0]);
                u32x4 g0;
                g0[0] = 1u;                                    // count=1
                g0[1] = ldsa;                                  // lds_addr
                g0[2] = (uint32_t)ga;                          // global_addr[31:0]
                g0[3] = (uint32_t)(ga >> 32) | (2u << 30);     // ga[56:32] | type=2
                u32x8 g1;
                g1[0] = (2u << 16) | (1u << 20) | (3u << 22);  // 4B, pad_en, 16DW/1DW
                g1[1] = 512u << 16;                            // tensor_dim0 = 512
                g1[2] = 512u << 16;                            // dim0_hi=0 | dim1 = 512
                g1[3] = 16u << 16;                             // dim1_hi=0 | tile_dim0=16
                g1[4] = 128u;                                  // tile_dim1=128, tile_dim2=0
                g1[5] = 512u;                                  // dim0_stride = 512
                g1[6] = 0u;
                g1[7] = 0u;
                asm volatile("tensor_load_to_lds %0, %1" :: "s"(g0), "s"(g1) : "memory");
                __builtin_amdgcn_s_wait_tensorcnt(0);
            }
#else
            for (int idx = tid; idx < 2048; idx += 256) {
                int m = idx >> 4, cch = idx & 15;
                Ah[m * 17 + cch] =
                    wpk[((size_t)tapf * 512 + oBase + m) * 512 + icB + cch];
            }
#endif
            __syncthreads();
            #pragma unroll
            for (int k4 = 0; k4 < 4; ++k4) {
                const int kk = k4 * 4 + khalf;
                v2f a0, a1, b0v, b1v;
                a0.x = Ah[(wm * 32 + mrow) * 17 + kk];
                a0.y = Ah[(wm * 32 + mrow) * 17 + kk + 1];
                a1.x = Ah[(wm * 32 + 16 + mrow) * 17 + kk];
                a1.y = Ah[(wm * 32 + 16 + mrow) * 17 + kk + 1];
                b0v.x = Bh[( kk      * 10 + py0 + ky) * 11 + px0 + kx];
                b0v.y = Bh[((kk + 1) * 10 + py0 + ky) * 11 + px0 + kx];
                b1v.x = Bh[( kk      * 10 + py1 + ky) * 11 + px1 + kx];
                b1v.y = Bh[((kk + 1) * 10 + py1 + ky) * 11 + px1 + kx];
                c00 = __builtin_amdgcn_wmma_f32_16x16x4_f32(false, a0, false, b0v,
                                                            (short)0, c00, false, false);
                c01 = __builtin_amdgcn_wmma_f32_16x16x4_f32(false, a0, false, b1v,
                                                            (short)0, c01, false, false);
                c10 = __builtin_amdgcn_wmma_f32_16x16x4_f32(false, a1, false, b0v,
                                                            (short)0, c10, false, false);
                c11 = __builtin_amdgcn_wmma_f32_16x16x4_f32(false, a1, false, b1v,
                                                            (short)0, c11, false, false);
            }
        }
    }

    const float ns = nsPtr[0];
    v8f accs[4] = {c00, c01, c10, c11};
    #pragma unroll
    for (int sub = 0; sub < 4; ++sub) {
        int subm = sub >> 1, subn = sub & 1;
        #pragma unroll
        for (int r = 0; r < 8; ++r) {
            // D(r,lane): m = r + (lane>=16)*8, n = lane&15  (ISA 7.12.2 C/D layout)
            int o  = oBase + wm * 32 + subm * 16 + r + (lane >> 4) * 8;
            int nn = wn * 32 + subn * 16 + (lane & 15);
            int gy = tileY * 8 + (nn >> 3);
            int gx = tileX * 8 + (nn & 7);
            float v = accs[sub][r] * dmod[b * 512 + o] + noise[gy * 64 + gx] * ns + bias[o];
            v = (v >= 0.f ? v : 0.2f * v) * LRELU_GAIN;
            v = fminf(fmaxf(v, -256.f), 256.f);
            out[(((size_t)b * 512 + o) * 64 + gy) * 64 + gx] = v;
        }
    }
}

// ---------------------------------------------------------------------------
// toRGB: y = clip(sum_i h*st*wt + bt) + upfirdn(img, up=2, pad 2/1, gain 4)
// ---------------------------------------------------------------------------
__global__ void torgb_kernel(const float* __restrict__ h,
                             const float* __restrict__ st,   // already * 1/sqrt(512)
                             const float* __restrict__ wt,   // (3,512)
                             const float* __restrict__ bt,   // (3)
                             const float* __restrict__ img,  // (8,3,32,32)
                             float* __restrict__ out2)       // (8,3,64,64)
{
    int idx = blockIdx.x * 256 + threadIdx.x;   // 0..32767
    int b = idx >> 12;
    int p = idx & 4095;
    int y = p >> 6, x = p & 63;
    const float* hp = h + (size_t)b * 512 * 4096 + p;
    const float* sp = st + (size_t)b * 512;
    float acc0 = 0.f, acc1 = 0.f, acc2 = 0.f;
    for (int i = 0; i < 512; ++i) {
        float hs = hp[(size_t)i * 4096] * sp[i];
        acc0 += hs * wt[i];
        acc1 += hs * wt[512 + i];
        acc2 += hs * wt[1024 + i];
    }
    const float F[4] = {0.125f, 0.375f, 0.375f, 0.125f};
    #pragma unroll
    for (int o = 0; o < 3; ++o) {
        float v = (o == 0 ? acc0 : o == 1 ? acc1 : acc2) + bt[o];
        v = fminf(fmaxf(v, -256.f), 256.f);
        float u = 0.f;
        #pragma unroll
        for (int ty = 0; ty < 4; ++ty) {
            int zr = y + ty - 2;
            if (zr < 0 || zr >= 64 || (zr & 1)) continue;
            int ir = zr >> 1;
            #pragma unroll
            for (int tx = 0; tx < 4; ++tx) {
                int zc = x + tx - 2;
                if (zc < 0 || zc >= 64 || (zc & 1)) continue;
                u += F[ty] * F[tx] * 4.0f * img[(((size_t)b * 3 + o) * 32 + ir) * 32 + (zc >> 1)];
            }
        }
        out2[(((size_t)b * 3 + o) * 64 + y) * 64 + x] = v + u;
    }
}

// ---------------------------------------------------------------------------
extern "C" void kernel_launch(void* const* d_in, const int* in_sizes, int n_in,
                              void* d_out, int out_size, void* d_ws, size_t ws_size,
                              hipStream_t stream)
{
    (void)in_sizes; (void)n_in; (void)out_size; (void)ws_size;
    const float* x    = (const float*)d_in[0];
    const float* img  = (const float*)d_in[1];
    const float* wsty = (const float*)d_in[2];
    const float* a0w  = (const float*)d_in[3];
    const float* a0b  = (const float*)d_in[4];
    const float* w0   = (const float*)d_in[5];
    const float* b0   = (const float*)d_in[6];
    const float* ns0  = (const float*)d_in[7];
    const float* nc0  = (const float*)d_in[8];
    const float* a1w  = (const float*)d_in[9];
    const float* a1b  = (const float*)d_in[10];
    const float* w1   = (const float*)d_in[11];
    const float* b1   = (const float*)d_in[12];
    const float* ns1  = (const float*)d_in[13];
    const float* nc1  = (const float*)d_in[14];
    const float* atw  = (const float*)d_in[15];
    const float* atb  = (const float*)d_in[16];
    const float* wt   = (const float*)d_in[17];
    const float* bt   = (const float*)d_in[18];

    // workspace layout (floats): ~22.1M floats (~88 MB)
    float* W    = (float*)d_ws;
    float* s0   = W;                    // 4096
    float* s1   = W + 4096;             // 4096
    float* st   = W + 8192;             // 4096
    float* dm0  = W + 12288;            // 4096
    float* dm1  = W + 16384;            // 4096
    float* wsq0 = W + 20480;            // 262144
    float* wsq1 = wsq0 + 262144;        // 262144
    float* wpk0 = wsq1 + 262144;        // 9*512*512 = 2359296
    float* wpk1 = wpk0 + 2359296;       // 2359296
    float* h0   = wpk1 + 2359296;       // 8*512*64*64 = 16777216

    float* hOut   = (float*)d_out;      // first output: h (8,512,64,64)
    float* imgOut = hOut + 16777216;    // second output: (8,3,64,64)

    affine_kernel<<<16, 256, 0, stream>>>(wsty, 0, a0w, a0b, 1.0f, s0);
    affine_kernel<<<16, 256, 0, stream>>>(wsty, 1, a1w, a1b, 1.0f, s1);
    affine_kernel<<<16, 256, 0, stream>>>(wsty, 2, atw, atb, INV_SQRT512, st);
    wsq_kernel<<<1024, 256, 0, stream>>>(w0, wsq0);
    wsq_kernel<<<1024, 256, 0, stream>>>(w1, wsq1);
    demod_kernel<<<16, 256, 0, stream>>>(s0, wsq0, dm0);
    demod_kernel<<<16, 256, 0, stream>>>(s1, wsq1, dm1);
    pack_w_kernel<<<9216, 256, 0, stream>>>(w0, wpk0);
    pack_w_kernel<<<9216, 256, 0, stream>>>(w1, wpk1);
    conv_wmma_kernel<true><<<dim3(64, 4, 8), 256, 0, stream>>>(
        x, wpk0, s0, dm0, nc0, ns0, b0, h0);
    conv_wmma_kernel<false><<<dim3(64, 4, 8), 256, 0, stream>>>(
        h0, wpk1, s1, dm1, nc1, ns1, b1, hOut);
    torgb_kernel<<<128, 256, 0, stream>>>(hOut, st, wt, bt, img, imgOut);
}